// TransNAS_64183991271927
// MI455X (gfx1250) — compile-verified
//
#include <hip/hip_runtime.h>
#include <hip/hip_bf16.h>

// TransNAS graph tokenizer for MI455X (gfx1250, wave32).
// Bandwidth-bound (~136MB out @ 23.3TB/s ~ 6us floor, only ~2.2 GFLOP fp32),
// so we stay in f32 and use V_WMMA_F32_16X16X4_F32 for the two GEMMs.

typedef __attribute__((ext_vector_type(2))) float v2f;
typedef __attribute__((ext_vector_type(8))) float v8f;

#define B_       64
#define N_       64
#define D_       256
#define LAP_     16
#define M_       2016      // N*(N-1)/2
#define SEQ_     2081      // 1 + N + M
#define NT_      16        // D/16 column tiles
#define MT_NODE_ 4         // N/16 row tiles
#define MT_EDGE_ 126       // M/16 row tiles

// ---------------------------------------------------------------------------
// Kernel 1: stable edge compaction. One wave32 per batch.
// Order = upper-triangular row-major; valid edges first (stable), matching
// jnp.argsort(~valid, stable=True). Packs (valid<<12)|(i<<6)|j per slot.
// ---------------------------------------------------------------------------
__device__ __forceinline__ void decode_triu(int m, int& i, int& j) {
    int ii = 0;
    while (m >= (N_ - 1 - ii)) { m -= (N_ - 1 - ii); ++ii; }
    i = ii;
    j = ii + 1 + m;
}

__global__ void compact_edges(const int* __restrict__ adj, int* __restrict__ codes) {
    const int b = blockIdx.x;
    const int lane = threadIdx.x;                 // 0..31 (wave32)
    const unsigned lt = (1u << lane) - 1u;

    // pass 1: total valid count
    int nv = 0;
    for (int c = 0; c < M_ / 32; ++c) {
        int i, j;
        decode_triu(c * 32 + lane, i, j);
        const int v = adj[b * N_ * N_ + i * N_ + j] > 0;
        const unsigned mask = __builtin_amdgcn_ballot_w32(v != 0);
        nv += __popc(mask);
    }
    // pass 2: scatter to compacted positions
    int vbase = 0, ibase = nv;
    for (int c = 0; c < M_ / 32; ++c) {
        int i, j;
        decode_triu(c * 32 + lane, i, j);
        const int v = adj[b * N_ * N_ + i * N_ + j] > 0;
        const unsigned mask = __builtin_amdgcn_ballot_w32(v != 0);
        const int vbelow = __popc(mask & lt);
        const int pos = v ? (vbase + vbelow) : (ibase + lane - vbelow);
        codes[b * M_ + pos] = (v << 12) | (i << 6) | j;
        const int cnt = __popc(mask);
        vbase += cnt;
        ibase += 32 - cnt;
    }
}

// ---------------------------------------------------------------------------
// Kernel 2: graph token row + mask zeros for [graph, nodes] slots.
// ---------------------------------------------------------------------------
__global__ void graph_token(const float* __restrict__ gtok, float* __restrict__ out) {
    const int b = blockIdx.x;
    const int d = threadIdx.x;                    // 0..255
    float* seq  = out;
    float* mask = out + (size_t)B_ * SEQ_ * D_;
    seq[(size_t)b * SEQ_ * D_ + d] = gtok[d];
    if (d < 1 + N_) mask[(size_t)b * SEQ_ + d] = 0.0f;
}

// ---------------------------------------------------------------------------
// Kernel 3: node tokens. One wave per 16x16 output tile.
// acc = node_feats_tile; acc += eigvec_tile(16x16) @ W_lap_tile(16x16)
// via 4x V_WMMA_F32_16X16X4_F32.
// ---------------------------------------------------------------------------
__global__ void node_tok_kernel(const float* __restrict__ feats,
                                const float* __restrict__ eig,
                                const float* __restrict__ Wlap,
                                float* __restrict__ out) {
    const int lane = threadIdx.x & 31;
    const int wave = threadIdx.x >> 5;
    int t = blockIdx.x * (blockDim.x >> 5) + wave;
    const int nt = t % NT_;      t /= NT_;
    const int mt = t % MT_NODE_; t /= MT_NODE_;
    const int b  = t;

    const int half = lane >> 4;       // 0: K/M pairs {0,1}; 1: {2,3} (A/B), M+8 (C/D)
    const int l16  = lane & 15;
    const int col  = nt * 16 + l16;
    const int r0   = mt * 16;

    // C init = node_feats tile (C/D layout: VGPR v -> row v + 8*half, col l16)
    v8f acc;
#pragma unroll
    for (int v = 0; v < 8; ++v)
        acc[v] = feats[(size_t)b * N_ * D_ + (size_t)(r0 + v + 8 * half) * D_ + col];

    const float* arow = eig + (size_t)b * N_ * LAP_ + (size_t)(r0 + l16) * LAP_;
#pragma unroll
    for (int kk = 0; kk < LAP_; kk += 4) {
        const int k = kk + 2 * half;
        v2f a;  a.x  = arow[k];              a.y  = arow[k + 1];
        v2f bm; bm.x = Wlap[k * D_ + col];   bm.y = Wlap[(k + 1) * D_ + col];
        acc = __builtin_amdgcn_wmma_f32_16x16x4_f32(false, a, false, bm,
                                                    (short)0, acc, false, false);
    }

    float* seq = out + (size_t)b * SEQ_ * D_;
#pragma unroll
    for (int v = 0; v < 8; ++v)
        seq[(size_t)(1 + r0 + v + 8 * half) * D_ + col] = acc[v];
}

// ---------------------------------------------------------------------------
// Kernel 4: edge tokens. One wave per 16x16 output tile.
// acc = Pu @ W_edge[1:17] + Pv @ W_edge[17:33]   (8x V_WMMA_F32_16X16X4_F32)
// final = valid ? acc + (W_edge[0] + b_edge + type_embed[1]) : 0
// ---------------------------------------------------------------------------
__global__ void edge_tok_kernel(const float* __restrict__ eig,
                                const float* __restrict__ Wedge,
                                const float* __restrict__ bedge,
                                const float* __restrict__ temb,
                                const int* __restrict__ codes,
                                float* __restrict__ out) {
    const int lane = threadIdx.x & 31;
    const int wave = threadIdx.x >> 5;
    int t = blockIdx.x * (blockDim.x >> 5) + wave;
    const int nt = t % NT_;      t /= NT_;
    const int mt = t % MT_EDGE_; t /= MT_EDGE_;
    const int b  = t;

    const int half = lane >> 4;
    const int l16  = lane & 15;
    const int col  = nt * 16 + l16;
    const int pos0 = mt * 16;

    const int* cbase = codes + b * M_;
    const int mycode = cbase[pos0 + l16];         // A-row assignment: row = l16
    const int si = (mycode >> 6) & 63;
    const int di = mycode & 63;
    const float* pu = eig + (size_t)b * N_ * LAP_ + (size_t)si * LAP_;
    const float* pv = eig + (size_t)b * N_ * LAP_ + (size_t)di * LAP_;

    v8f acc = {};
#pragma unroll
    for (int kk = 0; kk < LAP_; kk += 4) {        // Pu x W_edge[1:17]
        const int k = kk + 2 * half;
        v2f a;  a.x  = pu[k];                       a.y  = pu[k + 1];
        v2f bm; bm.x = Wedge[(1 + k) * D_ + col];   bm.y = Wedge[(2 + k) * D_ + col];
        acc = __builtin_amdgcn_wmma_f32_16x16x4_f32(false, a, false, bm,
                                                    (short)0, acc, false, false);
    }
#pragma unroll
    for (int kk = 0; kk < LAP_; kk += 4) {        // Pv x W_edge[17:33]
        const int k = kk + 2 * half;
        v2f a;  a.x  = pv[k];                       a.y  = pv[k + 1];
        v2f bm; bm.x = Wedge[(17 + k) * D_ + col];  bm.y = Wedge[(18 + k) * D_ + col];
        acc = __builtin_amdgcn_wmma_f32_16x16x4_f32(false, a, false, bm,
                                                    (short)0, acc, false, false);
    }

    // fused constant: W_edge[0,col] (the "1" input row) + b_edge + type_embed[1]
    const float cvec = Wedge[col] + bedge[col] + temb[D_ + col];

    float* seq = out + (size_t)b * SEQ_ * D_;
#pragma unroll
    for (int v = 0; v < 8; ++v) {
        const int row  = pos0 + v + 8 * half;
        const int code = cbase[row];
        const float val = ((code >> 12) & 1) ? (acc[v] + cvec) : 0.0f;
        seq[(size_t)(1 + N_ + row) * D_ + col] = val;
    }

    if (nt == 0 && lane < 16) {
        float* mask = out + (size_t)B_ * SEQ_ * D_ + (size_t)b * SEQ_;
        mask[1 + N_ + pos0 + lane] = ((mycode >> 12) & 1) ? 0.0f : 1.0f;
    }
}

// ---------------------------------------------------------------------------
extern "C" void kernel_launch(void* const* d_in, const int* in_sizes, int n_in,
                              void* d_out, int out_size, void* d_ws, size_t ws_size,
                              hipStream_t stream) {
    (void)in_sizes; (void)n_in; (void)out_size; (void)ws_size;

    const int*   adj   = (const int*)d_in[0];
    const float* feats = (const float*)d_in[1];
    const float* eig   = (const float*)d_in[2];
    const float* Wlap  = (const float*)d_in[3];
    const float* Wedge = (const float*)d_in[4];
    const float* bedge = (const float*)d_in[5];
    const float* temb  = (const float*)d_in[6];
    const float* gtok  = (const float*)d_in[7];
    float* out   = (float*)d_out;
    int*   codes = (int*)d_ws;                    // B_*M_ ints = 516 KB

    compact_edges<<<B_, 32, 0, stream>>>(adj, codes);
    graph_token<<<B_, 256, 0, stream>>>(gtok, out);

    const int node_waves = B_ * MT_NODE_ * NT_;   // 4096
    node_tok_kernel<<<node_waves / 8, 256, 0, stream>>>(feats, eig, Wlap, out);

    const int edge_waves = B_ * MT_EDGE_ * NT_;   // 129024
    edge_tok_kernel<<<edge_waves / 8, 256, 0, stream>>>(eig, Wedge, bedge, temb, codes, out);
}